// L1_ChamferLoss_19164144075464
// MI455X (gfx1250) — compile-verified
//
#include <hip/hip_runtime.h>
#include <hip/hip_bf16.h>
#include <math.h>

// CDNA5 wave32 WMMA fragment types
typedef __attribute__((ext_vector_type(2))) float v2f;   // A/B frag of 16x16x4 f32 WMMA
typedef __attribute__((ext_vector_type(8))) float v8f;   // C/D frag (16x16 f32)

#define WAVES_PER_BLOCK 8
#define ROWS_PER_WAVE   16
#define ROWS_PER_BLOCK  (WAVES_PER_BLOCK * ROWS_PER_WAVE)   // 128
#define BK              128                                 // staged target points per round
#define SUBTILES        (BK / 16)                           // 8 WMMAs per staging round

// One direction of Chamfer: for each point p in P (rows), min squared distance
// to any point q in Q (cols); accumulate sum of sqrt(min d) into *acc.
// d(m,n) = |p_m|^2 + |q_n|^2 - 2 p.q  computed as a single V_WMMA_F32_16X16X4_F32:
//   A[m] = (px,py,pz,1) ; B[:,n] = (-2qx,-2qy,-2qz,|q|^2) ; C[m,n] = |p_m|^2
//
// LDS trick: the staged float4 (-2qx,-2qy,-2qz,|q|^2) is exactly the two
// B-fragment register pairs {(K0,K1),(K2,K3)} back to back, so each lane
// fetches its half-dependent pair with a single ds_load_b64 at offset half*8
// (no cndmask selects in the hot loop).
__global__ void __launch_bounds__(256)
chamfer_min_dir(const float* __restrict__ P,   // [B,N,3]
                const float* __restrict__ Q,   // [B,M,3]
                int N, int M,
                float* __restrict__ acc)
{
    __shared__ float4 sQ[BK];                       // (-2qx,-2qy,-2qz,|q|^2)
    __shared__ float  sA2[ROWS_PER_BLOCK];          // |p|^2 per staged row
    __shared__ float  sAcc;

    const int tid   = threadIdx.x;
    const int lane  = tid & 31;
    const int wave  = tid >> 5;
    const int half  = lane >> 4;    // 0: K=0..1 / rows 0..7 ; 1: K=2..3 / rows 8..15
    const int nloc  = lane & 15;
    const int batch = blockIdx.y;

    const float* Pb = P + (size_t)batch * N * 3;
    const float* Qb = Q + (size_t)batch * M * 3;

    // ---- Load this wave's 16-row A tile (both halves read the same rows) ----
    const int rowBase = blockIdx.x * ROWS_PER_BLOCK + wave * ROWS_PER_WAVE;
    const int mrow    = min(rowBase + nloc, N - 1);
    const float ax = Pb[mrow * 3 + 0];
    const float ay = Pb[mrow * 3 + 1];
    const float az = Pb[mrow * 3 + 2];
    const float a2 = ax * ax + ay * ay + az * az;
    if (lane < 16) sA2[wave * ROWS_PER_WAVE + nloc] = a2;

    // A-matrix 16x4 f32 layout: lane m=lane%16; V0 holds K = 0 (half0) / 2 (half1),
    // V1 holds K = 1 (half0) / 3 (half1).  A[m] = (ax, ay, az, 1.0)
    v2f afrag;
    afrag.x = half ? az   : ax;
    afrag.y = half ? 1.0f : ay;

    if (tid == 0) sAcc = 0.0f;
    __syncthreads();

    // C fragment: C[r] = |p|^2 of row (r + 8*half), uniform across N (lanes)
    v8f cfrag;
#pragma unroll
    for (int r = 0; r < 8; ++r)
        cfrag[r] = sA2[wave * ROWS_PER_WAVE + r + 8 * half];

    float dmin[8];
#pragma unroll
    for (int r = 0; r < 8; ++r) dmin[r] = 3.0e38f;

    // per-lane B-fragment source: pair 'half' of point nloc within each subtile
    const float2* sQ2 = (const float2*)sQ;
    const int qidx = (nloc << 1) | half;            // float2 index within a subtile

    // ---- Sweep all M target points, BK per LDS staging round ----
    for (int k0 = 0; k0 < M; k0 += BK) {
        __syncthreads();
        if (tid < BK) {
            const int n = min(k0 + tid, M - 1);
            const float bx = Qb[n * 3 + 0];
            const float by = Qb[n * 3 + 1];
            const float bz = Qb[n * 3 + 2];
            sQ[tid] = make_float4(-2.0f * bx, -2.0f * by, -2.0f * bz,
                                  bx * bx + by * by + bz * bz);
        }
        __syncthreads();

        // Issue all B-fragment LDS loads for the chunk up front (one wait
        // covers the batch instead of a full LDS round-trip per WMMA).
        v2f bf[SUBTILES];
#pragma unroll
        for (int s = 0; s < SUBTILES; ++s) {
            const float2 q = sQ2[(s << 5) + qidx];  // 32 float2 per subtile
            bf[s].x = q.x;
            bf[s].y = q.y;
        }

#pragma unroll
        for (int s = 0; s < SUBTILES; ++s) {
            // d = A*B + C : full squared distances for a 16x16 tile
            v8f d = __builtin_amdgcn_wmma_f32_16x16x4_f32(
                false, afrag, false, bf[s], (short)0, cfrag, false, false);

            // min accumulate (clamp-to-0 commutes with min; hoisted to the end)
#pragma unroll
            for (int r = 0; r < 8; ++r) dmin[r] = fminf(dmin[r], d[r]);
        }
    }

    // ---- Min-reduce across the 16 lanes of each half (N direction) ----
#pragma unroll
    for (int off = 1; off < 16; off <<= 1) {
#pragma unroll
        for (int r = 0; r < 8; ++r) {
            const float o = __shfl_xor(dmin[r], off, 32);
            dmin[r] = fminf(dmin[r], o);
        }
    }

    // Lane 0 of each half owns rows (0..7)+8*half of this wave's tile
    if (nloc == 0) {
        float s = 0.0f;
#pragma unroll
        for (int r = 0; r < 8; ++r) {
            const int row = rowBase + r + 8 * half;
            if (row < N) s += sqrtf(fmaxf(dmin[r], 0.0f));
        }
        atomicAdd(&sAcc, s);
    }
    __syncthreads();
    if (tid == 0) atomicAdd(acc, sAcc);
}

__global__ void init_acc_kernel(float* ws) {
    ws[0] = 0.0f;
    ws[1] = 0.0f;
}

__global__ void finalize_kernel(const float* __restrict__ ws,
                                float* __restrict__ out,
                                float invBN, float invBM) {
    out[0] = 0.5f * (ws[0] * invBN + ws[1] * invBM);
}

extern "C" void kernel_launch(void* const* d_in, const int* in_sizes, int n_in,
                              void* d_out, int out_size, void* d_ws, size_t ws_size,
                              hipStream_t stream) {
    const float* a1 = (const float*)d_in[0];   // [B,N,3] fp32
    const float* a2 = (const float*)d_in[1];   // [B,M,3] fp32
    float* out = (float*)d_out;
    float* ws  = (float*)d_ws;                 // ws[0]: sum sqrt(dist1), ws[1]: sum sqrt(dist2)

    const int B = 8;                            // reference: B=8, D=3
    const int N = in_sizes[0] / (B * 3);
    const int M = in_sizes[1] / (B * 3);

    init_acc_kernel<<<1, 1, 0, stream>>>(ws);

    dim3 block(256);
    dim3 g1((N + ROWS_PER_BLOCK - 1) / ROWS_PER_BLOCK, B);
    dim3 g2((M + ROWS_PER_BLOCK - 1) / ROWS_PER_BLOCK, B);
    chamfer_min_dir<<<g1, block, 0, stream>>>(a1, a2, N, M, ws + 0);
    chamfer_min_dir<<<g2, block, 0, stream>>>(a2, a1, M, N, ws + 1);

    finalize_kernel<<<1, 1, 0, stream>>>(ws, out,
                                         1.0f / ((float)B * (float)N),
                                         1.0f / ((float)B * (float)M));
}